// LinearMap_54047868452897
// MI455X (gfx1250) — compile-verified
//
#include <hip/hip_runtime.h>

// CDNA5 / gfx1250, wave32. Streaming triangular batched GEMV via
// V_WMMA_F32_16X16X4_F32 (exact f32). Memory-bound: reads ~256MB (strict
// lower triangle of h) once with NT hints. Software-pipelined one 16-K
// chunk deep so each wave keeps ~2KB of h in flight (hides HBM latency).

typedef __attribute__((ext_vector_type(2))) float v2f;
typedef __attribute__((ext_vector_type(8))) float v8f;

#define DIM 1024
#define NBATCH 128

__device__ __forceinline__ v2f nt_load2(const float* p) {
  return __builtin_nontemporal_load((const v2f*)p);
}

__global__ __launch_bounds__(256) void iaf_wmma_kernel(
    const float* __restrict__ z, const float* __restrict__ h,
    float* __restrict__ out) {
  __shared__ float zsh[DIM];

  const int b   = blockIdx.y;          // batch
  const int tid = threadIdx.x;

  // Stage z[b][:] into LDS: 256 threads x float4 = 1024 floats, coalesced.
  {
    const float4* zrow = (const float4*)(z + (size_t)b * DIM);
    ((float4*)zsh)[tid] = zrow[tid];
  }
  __syncthreads();

  const unsigned lane = tid & 31u;
  const unsigned wave = tid >> 5;      // 0..7
  const unsigned half = lane >> 4;     // lane-half: supplies K+2*half in A layout
  const unsigned m    = lane & 15u;    // row within 16-row tile
  const int r   = blockIdx.x * 8 + wave;   // row tile 0..63
  const int row = r * 16 + (int)m;         // global row i (per batch)

  // Per-lane base into h (A operand): lane (m,half) supplies K = k+2*half {+0,+1}.
  const float* hrow = h + ((size_t)b << 20) + (size_t)row * DIM + 2u * half;

  v8f acc = {};  // 16x16 f32 accumulator; every column identical (B replicated)

  // ---- Pipelined strictly-lower tiles: 16 K per iteration (kdiag = 16*r). ----
  // Prologue loads chunk 0; each iteration prefetches chunk k0+16 and runs
  // WMMAs on the chunk loaded last iteration. The final prefetch (k0+16 ==
  // kdiag) is the diagonal tile, consumed by the masked epilogue below.
  const int kdiag = 16 * r;
  v2f a0 = nt_load2(hrow);
  v2f a1 = nt_load2(hrow + 4);
  v2f a2 = nt_load2(hrow + 8);
  v2f a3 = nt_load2(hrow + 12);
  for (int k0 = 0; k0 < kdiag; k0 += 16) {
    // Prefetch next chunk (always in-bounds: k0+16 <= kdiag <= 1008).
    const float* hn = hrow + k0 + 16;
    v2f n0 = nt_load2(hn);
    v2f n1 = nt_load2(hn + 4);
    v2f n2 = nt_load2(hn + 8);
    v2f n3 = nt_load2(hn + 12);
    // LDS broadcast reads of z (B operand, replicated across columns).
    const float* zp = zsh + k0 + 2u * half;
    v2f b0; b0.x = zp[0];  b0.y = zp[1];
    v2f b1; b1.x = zp[4];  b1.y = zp[5];
    v2f b2; b2.x = zp[8];  b2.y = zp[9];
    v2f b3; b3.x = zp[12]; b3.y = zp[13];
    acc = __builtin_amdgcn_wmma_f32_16x16x4_f32(false, a0, false, b0, (short)0, acc, false, false);
    acc = __builtin_amdgcn_wmma_f32_16x16x4_f32(false, a1, false, b1, (short)0, acc, false, false);
    acc = __builtin_amdgcn_wmma_f32_16x16x4_f32(false, a2, false, b2, (short)0, acc, false, false);
    acc = __builtin_amdgcn_wmma_f32_16x16x4_f32(false, a3, false, b3, (short)0, acc, false, false);
    a0 = n0; a1 = n1; a2 = n2; a3 = n3;   // rotate pipeline
  }

  // ---- Diagonal tile (already in a0..a3): mask j >= i, 4 masked K-steps. ----
  {
    const int hj = 2 * (int)half;
    const float* zp = zsh + kdiag + hj;
    v2f bz;
    a0.x = (0  + hj < (int)m) ? a0.x : 0.0f;
    a0.y = (1  + hj < (int)m) ? a0.y : 0.0f;
    bz.x = zp[0];  bz.y = zp[1];
    acc = __builtin_amdgcn_wmma_f32_16x16x4_f32(false, a0, false, bz, (short)0, acc, false, false);
    a1.x = (4  + hj < (int)m) ? a1.x : 0.0f;
    a1.y = (5  + hj < (int)m) ? a1.y : 0.0f;
    bz.x = zp[4];  bz.y = zp[5];
    acc = __builtin_amdgcn_wmma_f32_16x16x4_f32(false, a1, false, bz, (short)0, acc, false, false);
    a2.x = (8  + hj < (int)m) ? a2.x : 0.0f;
    a2.y = (9  + hj < (int)m) ? a2.y : 0.0f;
    bz.x = zp[8];  bz.y = zp[9];
    acc = __builtin_amdgcn_wmma_f32_16x16x4_f32(false, a2, false, bz, (short)0, acc, false, false);
    a3.x = (12 + hj < (int)m) ? a3.x : 0.0f;
    a3.y = (13 + hj < (int)m) ? a3.y : 0.0f;
    bz.x = zp[12]; bz.y = zp[13];
    acc = __builtin_amdgcn_wmma_f32_16x16x4_f32(false, a3, false, bz, (short)0, acc, false, false);
  }

  // Column N=0 of D lives in lanes with m==0: acc[v] = row (v + 8*half).
  if (m == 0) {
    float* orow = out + (size_t)b * DIM + r * 16 + 8 * (int)half;
    const float* zdiag = zsh + r * 16 + 8 * half;
    float4 lo, hi;   // + unit diagonal contribution z[i]
    lo.x = acc[0] + zdiag[0]; lo.y = acc[1] + zdiag[1];
    lo.z = acc[2] + zdiag[2]; lo.w = acc[3] + zdiag[3];
    hi.x = acc[4] + zdiag[4]; hi.y = acc[5] + zdiag[5];
    hi.z = acc[6] + zdiag[6]; hi.w = acc[7] + zdiag[7];
    ((float4*)orow)[0] = lo;
    ((float4*)orow)[1] = hi;
  }

  // logdet = 0 (unit triangular), single writer.
  if (b == 0 && blockIdx.x == 0 && tid == 0)
    out[(size_t)NBATCH * DIM] = 0.0f;
}

extern "C" void kernel_launch(void* const* d_in, const int* in_sizes, int n_in,
                              void* d_out, int out_size, void* d_ws, size_t ws_size,
                              hipStream_t stream) {
  (void)in_sizes; (void)n_in; (void)d_ws; (void)ws_size; (void)out_size;
  const float* z = (const float*)d_in[0];  // (128, 1024)
  const float* h = (const float*)d_in[1];  // (128, 1024*1024)
  float* out = (float*)d_out;              // 128*1024 f_z + 1 logdet

  dim3 grid(8, NBATCH);   // 8 row-tile groups x 128 batches
  dim3 block(256);        // 8 wave32 waves; each wave owns one 16-row tile
  iaf_wmma_kernel<<<grid, block, 0, stream>>>(z, h, out);
}